// CTCLoss_18107582120090
// MI455X (gfx1250) — compile-verified
//
#include <hip/hip_runtime.h>
#include <stdint.h>

// CTC forward (alpha) recursion for MI455X / gfx1250.
// T=1024, N=256, C=128, S=128, L=2S+1=257.
#define TT 1024
#define NN 256
#define CC 128
#define SS 128
#define LL 257
#define NEGF (-1e30f)
#define NTHR 288   // 9 wave32s; threads 0..256 own lattice positions

__global__ __launch_bounds__(NTHR)
void ctc_alpha_kernel(const float* __restrict__ lp,    // [T, N, C] log-probs
                      const int*   __restrict__ tgt,   // [N, S] labels in [1, C-1]
                      const int*   __restrict__ ilen,  // [N]
                      const int*   __restrict__ tlen,  // [N]
                      float*       __restrict__ loss_ws) // [N] per-sample loss/tlen
{
    // 8-slot ring of 512B rows, filled by the async global->LDS engine.
    __shared__ __align__(16) float lpb[8][CC];
    // alpha ping-pong, 2-entry NEG padding in front for branch-free l-1 / l-2.
    __shared__ float Abuf[2][LL + 2];

    const int n   = blockIdx.x;
    const int tid = threadIdx.x;
    const int l   = tid;

    // ---- per-thread static lattice data ----
    int  e  = 0;       // ext[l]: class index emitted at position l (0 = blank)
    bool sk = false;   // skip transition l-2 -> l allowed
    if (l < LL) {
        if (l & 1) {
            e = tgt[n * SS + (l >> 1)];
            if (l >= 3) {
                int ep = tgt[n * SS + (l >> 1) - 1];
                sk = (e != ep);          // e != 0 always (labels in [1,C-1])
            }
        }
    }
    const int mylen = ilen[n];

    // ---- alpha at t = 0 ----
    if (tid < 2) { Abuf[0][tid] = NEGF; Abuf[1][tid] = NEGF; }  // padding
    if (l < LL) {
        const float* lp0 = lp + (size_t)n * CC;   // row t=0
        float a = NEGF;
        if (l == 0)      a = lp0[0];
        else if (l == 1) a = lp0[e];
        Abuf[0][l + 2] = a;
    }

    // ---- prefetch rows 1..3 via async global->LDS (wave 0 only) ----
    const unsigned lane    = (unsigned)tid;                       // 0..31 in wave 0
    const unsigned ldsbase = (unsigned)(uintptr_t)(&lpb[0][0]);   // LDS byte offset
    if (tid < 32) {
        for (int r = 1; r <= 3; ++r) {
            unsigned long long ga = (unsigned long long)(uintptr_t)
                (lp + ((size_t)r * NN + n) * CC) + lane * 16u;
            unsigned la = ldsbase + (unsigned)((r & 7) * CC * 4) + lane * 16u;
            asm volatile("global_load_async_to_lds_b128 %0, %1, off"
                         :: "v"(la), "v"(ga) : "memory");
        }
    }

    // ---- sequential scan over t = 1..T-1, one barrier per step ----
    int cur = 0;
    for (int t = 1; t < TT; ++t) {
        if (tid < 32) {
            const int tp = t + 3;                  // keep 3 rows in flight
            if (tp < TT) {
                unsigned long long ga = (unsigned long long)(uintptr_t)
                    (lp + ((size_t)tp * NN + n) * CC) + lane * 16u;
                unsigned la = ldsbase + (unsigned)((tp & 7) * CC * 4) + lane * 16u;
                asm volatile("global_load_async_to_lds_b128 %0, %1, off"
                             :: "v"(la), "v"(ga) : "memory");
                asm volatile("s_wait_asynccnt 0x3" ::: "memory"); // row t landed
            } else {
                asm volatile("s_wait_asynccnt 0x0" ::: "memory"); // drain tail
            }
        }
        __syncthreads();   // row t visible in LDS; alpha[cur] fully written

        if (l < LL) {
            const float* Ac = Abuf[cur];
            float a0 = Ac[l + 2];
            float a1 = Ac[l + 1];
            float a2 = sk ? Ac[l] : NEGF;
            float emit = lpb[t & 7][e];
            float m  = fmaxf(a0, fmaxf(a1, a2));
            float s  = __expf(a0 - m) + __expf(a1 - m) + __expf(a2 - m);
            float nv = emit + m + __logf(s);
            nv = (t < mylen) ? nv : a0;            // freeze past input length
            Abuf[cur ^ 1][l + 2] = nv;
        }
        cur ^= 1;
    }
    __syncthreads();

    if (tid == 0) {
        const int tl = tlen[n];
        const float* Ac = Abuf[cur];
        float aL = Ac[2 * tl + 2];                 // final blank position
        float aP = Ac[2 * tl + 1];                 // last label position
        float m  = fmaxf(aL, aP);
        float loss = -(m + __logf(__expf(aL - m) + __expf(aP - m)));
        loss_ws[n] = loss / (float)tl;
    }
}

// Deterministic fixed-order mean over the N per-sample losses.
__global__ __launch_bounds__(NN)
void ctc_reduce_kernel(const float* __restrict__ ws, float* __restrict__ out)
{
    __shared__ float r[NN];
    const int tid = threadIdx.x;
    r[tid] = ws[tid];
    __syncthreads();
    for (int ofs = NN / 2; ofs > 0; ofs >>= 1) {
        if (tid < ofs) r[tid] += r[tid + ofs];
        __syncthreads();
    }
    if (tid == 0) out[0] = r[0] / (float)NN;
}

extern "C" void kernel_launch(void* const* d_in, const int* in_sizes, int n_in,
                              void* d_out, int out_size, void* d_ws, size_t ws_size,
                              hipStream_t stream)
{
    const float* lp   = (const float*)d_in[0];   // log_probs [T,N,C] f32
    const int*   tgt  = (const int*)  d_in[1];   // targets   [N,S]   i32
    const int*   ilen = (const int*)  d_in[2];   // input_lengths  [N] i32
    const int*   tlen = (const int*)  d_in[3];   // target_lengths [N] i32
    float*       ws   = (float*)d_ws;            // [N] per-sample losses

    ctc_alpha_kernel<<<NN, NTHR, 0, stream>>>(lp, tgt, ilen, tlen, ws);
    ctc_reduce_kernel<<<1, NN, 0, stream>>>(ws, (float*)d_out);
}